// Decoder_56203942035661
// MI455X (gfx1250) — compile-verified
//
#include <hip/hip_runtime.h>
#include <stdint.h>

#define NB   256            // graphs
#define SXN  256            // encoder nodes per graph
#define SYN  70             // decoder nodes per graph
#define FDIM 512
#define EMB  256
#define NY   (NB * SYN)     // 17920 decoder nodes
#define NE   71680          // edges
#define NSRT 8

#define KB   16             // K-slab staged in LDS
#define NBLK 64             // N tile per block
// 80-float (320B) row stride: keeps every LDS row 16B-aligned for async b128
// staging, and puts rows kk / kk+2 32 banks apart -> conflict-free fragments.
#define LDSP 80

typedef __attribute__((ext_vector_type(2))) float v2f;
typedef __attribute__((ext_vector_type(8))) float v8f;
typedef __attribute__((ext_vector_type(4))) int   v4i;

#define AS1 __attribute__((address_space(1)))
#define AS3 __attribute__((address_space(3)))

#if defined(__has_builtin)
#if __has_builtin(__builtin_amdgcn_global_load_async_to_lds_b128)
#define USE_ASYNC 1
#endif
#if __has_builtin(__builtin_amdgcn_s_wait_asynccnt)
#define HAVE_WAIT_ASYNC_BUILTIN 1
#endif
#endif

static __device__ __forceinline__ void async_wait0() {
#if defined(USE_ASYNC)
#if defined(HAVE_WAIT_ASYNC_BUILTIN)
  __builtin_amdgcn_s_wait_asynccnt(0);
#else
  asm volatile("s_wait_asynccnt 0x0" ::: "memory");
#endif
#endif
}

static __device__ __forceinline__ void copy16_g2lds(const float* g, float* l) {
#if defined(USE_ASYNC)
  AS1 void* gv = (AS1 void*)(void*)const_cast<float*>(g);
  AS3 void* lv = (AS3 void*)(void*)l;
  __builtin_amdgcn_global_load_async_to_lds_b128((AS1 v4i*)gv, (AS3 v4i*)lv, 0, 0);
#else
  *(float4*)l = *(const float4*)g;
#endif
}

// ---------------------------------------------------------------------------
// fp32 WMMA GEMM:  C[M,N] = A[M,K] @ B  (+ bias)
//   BT == 0 : B stored row-major [K,N]
//   BT == 1 : B stored row-major [N,K]  (multiply by B^T; transposed at staging)
// Batched via sA/sB/sC element strides, blockIdx.z = batch.
// 4 waves / block; block computes 64(M) x 64(N); B slab (16 x 64) staged in
// LDS, double buffered (async-to-LDS when available). Requires N%64==0,
// K%16==0 (true for every call here); ragged M handled by clamp + guards.
// ---------------------------------------------------------------------------
template <int BT>
__global__ void __launch_bounds__(128)
wmma_gemm(const float* __restrict__ A, const float* __restrict__ B,
          float* __restrict__ C, const float* __restrict__ bias,
          int M, int N, int K, int lda, int ldb, int ldc,
          long sA, long sB, long sC) {
  __shared__ float Bs[2][KB][LDSP];

  const int tid  = threadIdx.x;
  const int lane = tid & 31;
  const int wv   = tid >> 5;
  const long z = blockIdx.z;
  A += z * sA; B += z * sB; C += z * sC;

  const int m0 = blockIdx.y * 64 + wv * 16;
  const int n0 = blockIdx.x * NBLK;

  int mrow = m0 + (lane & 15);
  mrow = (mrow < M) ? mrow : (M - 1);       // clamp ragged M (rows independent)
  const int kh = (lane >> 4) * 2;           // 0 or 2
  const int nc = lane & 15;

  // Stage one 16 x 64 slab of the mathematical B matrix into Bs[buf].
  auto stage = [&](int buf, int k0) {
    if (BT) {
      // B stored [N,K]: transpose while staging (8 k per thread along a row n)
      const int nr = tid & 63;
      const int kb = (tid >> 6) * 8;
      const float* gp = B + (long)(n0 + nr) * ldb + (k0 + kb);
      const float4 v0 = *(const float4*)gp;
      const float4 v1 = *(const float4*)(gp + 4);
      float* col = &Bs[buf][kb][nr];
      col[0 * LDSP] = v0.x; col[1 * LDSP] = v0.y;
      col[2 * LDSP] = v0.z; col[3 * LDSP] = v0.w;
      col[4 * LDSP] = v1.x; col[5 * LDSP] = v1.y;
      col[6 * LDSP] = v1.z; col[7 * LDSP] = v1.w;
    } else {
      // B stored [K,N]: direct copy (async-to-LDS when available)
      const int r  = tid >> 3;
      const int cb = (tid & 7) * 8;
      const float* gp = B + (long)(k0 + r) * ldb + n0 + cb;
      copy16_g2lds(gp,     &Bs[buf][r][cb]);
      copy16_g2lds(gp + 4, &Bs[buf][r][cb + 4]);
    }
  };

  v8f acc0 = {}, acc1 = {}, acc2 = {}, acc3 = {};

  stage(0, 0);
  if (!BT) async_wait0();
  __syncthreads();

  int buf = 0;
  for (int k0 = 0; k0 < K; k0 += KB) {
    if (k0 + KB < K) stage(buf ^ 1, k0 + KB);

    // prefetch all A fragments for this slab (4 substeps)
    const float* ap = A + (long)mrow * lda + k0 + kh;
    const v2f a0 = *(const v2f*)(ap);
    const v2f a1 = *(const v2f*)(ap + 4);
    const v2f a2 = *(const v2f*)(ap + 8);
    const v2f a3 = *(const v2f*)(ap + 12);
    const v2f af[4] = {a0, a1, a2, a3};

#pragma unroll
    for (int s = 0; s < 4; ++s) {
      const v2f a = af[s];
      const float* bp = &Bs[buf][4 * s + kh][nc];
      v2f b0, b1, b2, b3;
      b0.x = bp[0];  b0.y = bp[LDSP];
      b1.x = bp[16]; b1.y = bp[LDSP + 16];
      b2.x = bp[32]; b2.y = bp[LDSP + 32];
      b3.x = bp[48]; b3.y = bp[LDSP + 48];
      acc0 = __builtin_amdgcn_wmma_f32_16x16x4_f32(false, a, false, b0, (short)0, acc0, false, false);
      acc1 = __builtin_amdgcn_wmma_f32_16x16x4_f32(false, a, false, b1, (short)0, acc1, false, false);
      acc2 = __builtin_amdgcn_wmma_f32_16x16x4_f32(false, a, false, b2, (short)0, acc2, false, false);
      acc3 = __builtin_amdgcn_wmma_f32_16x16x4_f32(false, a, false, b3, (short)0, acc3, false, false);
    }

    if (!BT) async_wait0();
    __syncthreads();
    buf ^= 1;
  }

  const int rbase = m0 + (lane >> 4) * 8;
  const int c = n0 + nc;
#pragma unroll
  for (int i = 0; i < 8; ++i) {
    const int row = rbase + i;
    if (row >= M) continue;
    float* cp = C + (long)row * ldc;
    if (c      < N) cp[c     ] = acc0[i] + (bias ? bias[c     ] : 0.f);
    if (c + 16 < N) cp[c + 16] = acc1[i] + (bias ? bias[c + 16] : 0.f);
    if (c + 32 < N) cp[c + 32] = acc2[i] + (bias ? bias[c + 32] : 0.f);
    if (c + 48 < N) cp[c + 48] = acc3[i] + (bias ? bias[c + 48] : 0.f);
  }
}

// ---------------------------------------------------------------------------
// In-place softmax over rows of 256 (scale applied first). Wave per row.
// ---------------------------------------------------------------------------
__global__ void softmax256(float* __restrict__ data, int rows, float scale) {
  const int lane = threadIdx.x & 31;
  const long r = (long)blockIdx.x * (blockDim.x >> 5) + (threadIdx.x >> 5);
  if (r >= rows) return;
  float* row = data + r * 256;
  float v[8];
  float mx = -3.402823466e38f;
#pragma unroll
  for (int j = 0; j < 8; ++j) { v[j] = row[lane + 32 * j] * scale; mx = fmaxf(mx, v[j]); }
#pragma unroll
  for (int off = 16; off > 0; off >>= 1) mx = fmaxf(mx, __shfl_xor(mx, off, 32));
  float s = 0.f;
#pragma unroll
  for (int j = 0; j < 8; ++j) { v[j] = __expf(v[j] - mx); s += v[j]; }
#pragma unroll
  for (int off = 16; off > 0; off >>= 1) s += __shfl_xor(s, off, 32);
  const float inv = 1.f / s;
#pragma unroll
  for (int j = 0; j < 8; ++j) row[lane + 32 * j] = v[j] * inv;
}

// ---------------------------------------------------------------------------
// agg[tgt[e]] += h[src[e]] + Ee[etype[e]]   (wave per edge, f32 atomics)
// ---------------------------------------------------------------------------
__global__ void scatter_edges(const float* __restrict__ h, const int* __restrict__ src,
                              const int* __restrict__ tgt, const int* __restrict__ et,
                              const float* __restrict__ Ee, float* __restrict__ agg) {
  const long t = (long)blockIdx.x * blockDim.x + threadIdx.x;
  const long e = t >> 5;
  const int lane = t & 31;
  if (e >= NE) return;
  const int s = src[e], d = tgt[e], ty = et[e];
  const float* hp = h + (long)s * EMB;
  const float* ep = Ee + (long)ty * EMB;
  float* ap = agg + (long)d * EMB;
#pragma unroll
  for (int j = 0; j < 8; ++j) {
    const int ch = lane + 32 * j;
    atomicAdd(ap + ch, hp[ch] + ep[ch]);
  }
}

__global__ void fill_zero(float* __restrict__ p, long n) {
  const long i = (long)blockIdx.x * blockDim.x + threadIdx.x;
  if (i < n) p[i] = 0.f;
}

__global__ void gather_embed(const float* __restrict__ emb, const int* __restrict__ ids,
                             float* __restrict__ y) {
  const long i = (long)blockIdx.x * blockDim.x + threadIdx.x;
  if (i >= (long)NY * EMB) return;
  y[i] = emb[(long)ids[i >> 8] * EMB + (i & 255)];
}

__global__ void combine_relu(const float* __restrict__ h, const float* __restrict__ agg,
                             const float* __restrict__ c, float* __restrict__ y,
                             float* __restrict__ y2) {
  const long i = (long)blockIdx.x * blockDim.x + threadIdx.x;
  if (i >= (long)NY * EMB) return;
  float v = h[i] + agg[i] + c[i];
  v = v > 0.f ? v : 0.f;
  y[i] = v;
  if (y2) y2[i] = v;
}

// ---------------------------------------------------------------------------
// out[e,0:8] = concat(y[src[e]], y[tgt[e]]) @ W[512,8] + b   (wave per edge)
// ---------------------------------------------------------------------------
__global__ void edge_scores(const float* __restrict__ y, const int* __restrict__ src,
                            const int* __restrict__ tgt, const float* __restrict__ W,
                            const float* __restrict__ bias, float* __restrict__ out) {
  const long t = (long)blockIdx.x * blockDim.x + threadIdx.x;
  const long e = t >> 5;
  const int lane = t & 31;
  if (e >= NE) return;
  const int n = (lane < 16) ? src[e] : tgt[e];
  const float* row = y + (long)n * EMB + (lane & 15) * 16;
  const int wrow = lane * 16;  // concat channel base (0..496)
  float p[8] = {0.f, 0.f, 0.f, 0.f, 0.f, 0.f, 0.f, 0.f};
#pragma unroll
  for (int j = 0; j < 16; ++j) {
    const float v = row[j];
    const float* wr = W + (long)(wrow + j) * NSRT;
#pragma unroll
    for (int o = 0; o < NSRT; ++o) p[o] += v * wr[o];
  }
#pragma unroll
  for (int o = 0; o < NSRT; ++o)
#pragma unroll
    for (int off = 16; off > 0; off >>= 1) p[o] += __shfl_xor(p[o], off, 32);
  if (lane == 0) {
#pragma unroll
    for (int o = 0; o < NSRT; ++o) out[e * NSRT + o] = p[o] + bias[o];
  }
}

// ---------------------------------------------------------------------------
extern "C" void kernel_launch(void* const* d_in, const int* in_sizes, int n_in,
                              void* d_out, int out_size, void* d_ws, size_t ws_size,
                              hipStream_t stream) {
  const float* x      = (const float*)d_in[0];
  const int*   tgt_y  = (const int*)d_in[2];
  const int*   eidx   = (const int*)d_in[3];
  const int*   etype  = (const int*)d_in[4];
  const float* embeds = (const float*)d_in[6];
  const float* W_in[3] = {(const float*)d_in[7],  (const float*)d_in[11], (const float*)d_in[15]};
  const float* Wq[3]   = {(const float*)d_in[8],  (const float*)d_in[12], (const float*)d_in[16]};
  const float* Wv[3]   = {(const float*)d_in[9],  (const float*)d_in[13], (const float*)d_in[17]};
  const float* Ee[3]   = {(const float*)d_in[10], (const float*)d_in[14], (const float*)d_in[18]};
  const float* lin_z_w = (const float*)d_in[19];
  const float* lin_z_b = (const float*)d_in[20];
  const float* lin_g_w = (const float*)d_in[21];
  const float* lin_g_b = (const float*)d_in[22];

  const int* src  = eidx;        // tgt_edge_index[0]
  const int* tgtn = eidx + NE;   // tgt_edge_index[1]

  // Output regions (return order: y, y_score, y_edge_rel_score, a1, a2, a3)
  float* out       = (float*)d_out;
  float* out_y     = out;
  float* out_score = out_y + (long)NY * EMB;
  float* out_edge  = out_score + (long)NY * EMB;
  float* out_a[3];
  out_a[0] = out_edge + (long)NE * NSRT;
  out_a[1] = out_a[0] + (long)NY * SXN;
  out_a[2] = out_a[1] + (long)NY * SXN;

  // Workspace layout (floats)
  float* ws    = (float*)d_ws;
  float* y_cur = ws;                          // NY*EMB
  float* h     = y_cur + (long)NY * EMB;      // NY*EMB
  float* agg   = h     + (long)NY * EMB;      // NY*EMB
  float* ctxv  = agg   + (long)NY * EMB;      // NY*EMB
  float* q     = ctxv  + (long)NY * EMB;      // NY*FDIM (reused as ctx)
  float* ctx   = q;                           // alias: q dead after logits

  const dim3 blk(128);
  auto gemm = [&](int BT, const float* A, const float* B, float* C, const float* bias,
                  int M, int N, int K, int lda, int ldb, int ldc,
                  long sA, long sB, long sC, int batch) {
    dim3 grid((N + NBLK - 1) / NBLK, (M + 63) / 64, batch);
    if (BT) wmma_gemm<1><<<grid, blk, 0, stream>>>(A, B, C, bias, M, N, K, lda, ldb, ldc, sA, sB, sC);
    else    wmma_gemm<0><<<grid, blk, 0, stream>>>(A, B, C, bias, M, N, K, lda, ldb, ldc, sA, sB, sC);
  };

  const long nyemb = (long)NY * EMB;
  const int  TB = 256;
  const float inv_sqrt_f = 0.044194173824159216f;  // 1/sqrt(512)

  // y0 = embeds[tgt_y]
  gather_embed<<<(int)((nyemb + TB - 1) / TB), TB, 0, stream>>>(embeds, tgt_y, y_cur);

  for (int l = 0; l < 3; ++l) {
    // h = y @ W_in
    gemm(0, y_cur, W_in[l], h, nullptr, NY, EMB, EMB, EMB, EMB, EMB, 0, 0, 0, 1);
    // agg = segment_sum(h[src] + Ee[etype], tgt)
    fill_zero<<<(int)((nyemb + TB - 1) / TB), TB, 0, stream>>>(agg, nyemb);
    scatter_edges<<<(NE * 32 + TB - 1) / TB, TB, 0, stream>>>(h, src, tgtn, etype, Ee[l], agg);
    // q = h @ Wq   [NY, FDIM]
    gemm(0, h, Wq[l], q, nullptr, NY, FDIM, EMB, EMB, FDIM, FDIM, 0, 0, 0, 1);
    // logits = q_b @ x_b^T  -> alpha region  [B, SY, SX]
    gemm(1, q, x, out_a[l], nullptr, SYN, SXN, FDIM, FDIM, FDIM, SXN,
         (long)SYN * FDIM, (long)SXN * FDIM, (long)SYN * SXN, NB);
    // alpha = softmax(logits / sqrt(F))
    softmax256<<<(NY + 7) / 8, 256, 0, stream>>>(out_a[l], NY, inv_sqrt_f);
    // ctx = alpha @ x_b   [B, SY, FDIM]
    gemm(0, out_a[l], x, ctx, nullptr, SYN, FDIM, SXN, SXN, FDIM, FDIM,
         (long)SYN * SXN, (long)SXN * FDIM, (long)SYN * FDIM, NB);
    // ctxv = ctx @ Wv   [NY, EMB]
    gemm(0, ctx, Wv[l], ctxv, nullptr, NY, EMB, FDIM, FDIM, EMB, EMB, 0, 0, 0, 1);
    // y = relu(h + agg + ctxv)
    combine_relu<<<(int)((nyemb + TB - 1) / TB), TB, 0, stream>>>(
        h, agg, ctxv, y_cur, (l == 2) ? out_y : nullptr);
  }

  // y_score = y @ lin_z_w + lin_z_b
  gemm(0, y_cur, lin_z_w, out_score, lin_z_b, NY, 256, EMB, EMB, 256, 256, 0, 0, 0, 1);
  // y_edge_rel_score = concat(y[src], y[tgt]) @ lin_g_w + lin_g_b
  edge_scores<<<(NE * 32 + TB - 1) / TB, TB, 0, stream>>>(y_cur, src, tgtn, lin_g_w, lin_g_b, out_edge);
}